// DiBSFixed_88983132438713
// MI455X (gfx1250) — compile-verified
//
#include <hip/hip_runtime.h>
#include <math.h>

// ---------------- CDNA5 WMMA f32 16x16x4 ----------------
typedef __attribute__((ext_vector_type(2))) float v2f;
typedef __attribute__((ext_vector_type(8))) float v8f;

__device__ __forceinline__ v8f wmma4(v2f a, v2f b, v8f c) {
  // 8 args: (neg_a, A, neg_b, B, c_mod, C, reuse_a, reuse_b)
  return __builtin_amdgcn_wmma_f32_16x16x4_f32(false, a, false, b, (short)0, c,
                                               false, false);
}

#define DD 128
#define KLV 32
#define KS 64
#define NPTS 8192
#define INV_S2 100.0f      // 1/sigma_obs^2
#define C_OBS  1.3836465597893563f   // -0.5*log(2*pi*0.01)
#define C_TH  (-0.9189385332046727f) // -0.5*log(2*pi)
#define C_Z   (-0.9189385332046727f)

// A-matrix (16x4, M x K), row-major source with stride 128.
// lane 0-15: M=lane, VGPR0=K0, VGPR1=K1 ; lane 16-31: VGPR0=K2, VGPR1=K3
__device__ __forceinline__ v2f loadA128(const float* S, int mt, int k0, int lane) {
  int r = mt * 16 + (lane & 15);
  int k = k0 + ((lane >> 4) << 1);
  v2f a; a.x = S[r * 128 + k]; a.y = S[r * 128 + k + 1]; return a;
}
// B-matrix (4x16, K x N), row-major source with stride 128.
__device__ __forceinline__ v2f loadB128(const float* S, int nt, int k0, int lane) {
  int c = nt * 16 + (lane & 15);
  int k = k0 + ((lane >> 4) << 1);
  v2f b; b.x = S[k * 128 + c]; b.y = S[(k + 1) * 128 + c]; return b;
}

__device__ __forceinline__ float block_reduce(float v, float* scratch) {
  int tid = threadIdx.x;
  __syncthreads();               // protect scratch re-use
  scratch[tid] = v;
  __syncthreads();
  for (int s = 128; s > 0; s >>= 1) {
    if (tid < s) scratch[tid] += scratch[tid + s];
    __syncthreads();
  }
  float r = scratch[0];
  __syncthreads();
  return r;
}

// In-place 128x128 square: M <- M @ M  (256 threads = 8 waves, wave w = tile-row)
__device__ void square128_inplace(float* M) {
  int tid = threadIdx.x, lane = tid & 31, w = tid >> 5;
  v8f acc[8];
#pragma unroll
  for (int nt = 0; nt < 8; ++nt)
    acc[nt] = v8f{0.f, 0.f, 0.f, 0.f, 0.f, 0.f, 0.f, 0.f};
  for (int k0 = 0; k0 < 128; k0 += 4) {
    v2f fa = loadA128(M, w, k0, lane);
#pragma unroll
    for (int nt = 0; nt < 8; ++nt) {
      v2f fb = loadB128(M, nt, k0, lane);
      acc[nt] = wmma4(fa, fb, acc[nt]);
    }
  }
  __syncthreads();               // all reads of M done before overwrite
  int n0 = lane & 15, mb = (lane >> 4) ? 8 : 0;
#pragma unroll
  for (int nt = 0; nt < 8; ++nt)
#pragma unroll
    for (int r = 0; r < 8; ++r)
      M[(w * 16 + mb + r) * 128 + nt * 16 + n0] = acc[nt][r];
  __syncthreads();
}

// ---------------- Kernel 1: g_soft = sigmoid(alpha*(u v^T)*mask) ----------------
__global__ void k_gsoft(const float* __restrict__ z, float* __restrict__ gsoft,
                        float* __restrict__ out_tail) {
  int e = blockIdx.x * 256 + threadIdx.x;        // 16384 elems
  int i = e >> 7, j = e & 127;
  float dot = 0.f;
  for (int m = 0; m < KLV; ++m)
    dot += z[(i * KLV + m) * 2 + 0] * z[(j * KLV + m) * 2 + 1];
  float raw = (i == j) ? 0.f : 0.1f * dot;       // ALPHA=0.1, masked diag
  float g = 1.f / (1.f + __expf(-raw));
  gsoft[e] = g;
  out_tail[e] = g;
}

// ---------------- Kernel 2: XtX = x^T x  (WMMA, K=8192 split over 8 waves) ----
__global__ void k_xtx(const float* __restrict__ x, float* __restrict__ xtx) {
  __shared__ float part[8 * 256];
  int tid = threadIdx.x, lane = tid & 31, w = tid >> 5;
  int it = blockIdx.x >> 3, jt = blockIdx.x & 7;
  v8f acc = v8f{0.f, 0.f, 0.f, 0.f, 0.f, 0.f, 0.f, 0.f};
  int ca = it * 16 + (lane & 15);
  int cb = jt * 16 + (lane & 15);
  int khi = (lane >> 4) << 1;
  int nend = (w + 1) * 1024;
  for (int n0 = w * 1024; n0 < nend; n0 += 4) {
    int nb = n0 + khi;
    v2f fa; fa.x = x[nb * 128 + ca]; fa.y = x[(nb + 1) * 128 + ca];
    v2f fb; fb.x = x[nb * 128 + cb]; fb.y = x[(nb + 1) * 128 + cb];
    acc = wmma4(fa, fb, acc);
  }
  int n0l = lane & 15, mb = (lane >> 4) ? 8 : 0;
#pragma unroll
  for (int r = 0; r < 8; ++r) part[w * 256 + (mb + r) * 16 + n0l] = acc[r];
  __syncthreads();
  float s = 0.f;
#pragma unroll
  for (int w2 = 0; w2 < 8; ++w2) s += part[w2 * 256 + tid];
  int r = tid >> 4, c = tid & 15;
  xtx[(it * 16 + r) * 128 + jt * 16 + c] = s;
}

// ---------------- Kernel 3: per-sample k (grads_t, vals, h_k) ----------------
__global__ void k_perk(const float* __restrict__ theta, const float* __restrict__ unif,
                       const float* __restrict__ gsoft, const float* __restrict__ xtx,
                       float* __restrict__ grads, float* __restrict__ vals,
                       float* __restrict__ hk) {
  __shared__ float M[16384];                      // exactly 64 KB, multi-purpose
  int k = blockIdx.x;
  int tid = threadIdx.x, lane = tid & 31, w = tid >> 5;
  const float* U = unif + (size_t)k * 16384;

  // W = theta .* hard  into LDS
  for (int e = tid; e < 16384; e += 256) {
    float hard = (U[e] < gsoft[e]) ? 1.f : 0.f;
    M[e] = theta[e] * hard;
  }
  __syncthreads();

  // P = XtX @ W (WMMA); emit grads_t[k] and log-density partials on the fly
  float t1 = 0.f, t2 = 0.f, t3 = 0.f;
  int n0l = lane & 15, mb = (lane >> 4) ? 8 : 0;
  for (int nt = 0; nt < 8; ++nt) {
    v8f acc = v8f{0.f, 0.f, 0.f, 0.f, 0.f, 0.f, 0.f, 0.f};
    for (int k0 = 0; k0 < 128; k0 += 4) {
      v2f fa = loadA128(xtx, w, k0, lane);        // global (L2-resident)
      v2f fb = loadB128(M, nt, k0, lane);         // LDS
      acc = wmma4(fa, fb, acc);
    }
#pragma unroll
    for (int r = 0; r < 8; ++r) {
      int e = (w * 16 + mb + r) * 128 + nt * 16 + n0l;
      float Wij = M[e];
      float Xij = xtx[e];
      float hard = (U[e] < gsoft[e]) ? 1.f : 0.f;
      float P = acc[r];
      grads[(size_t)k * 16384 + e] = (Xij - P) * INV_S2 * hard - Wij;
      t1 += Xij * Wij;       // = tr(XtX W)   (XtX symmetric)
      t2 += Wij * P;         // = sum pred^2
      t3 += Wij * Wij;       // prior term
    }
  }
  // softmax-relevant part of log-density (common constants cancel)
  float T1 = block_reduce(t1, M);
  float T2 = block_reduce(t2, M);
  float T3 = block_reduce(t3, M);
  if (tid == 0) vals[k] = INV_S2 * T1 - 0.5f * INV_S2 * T2 - 0.5f * T3;

  // acyclicity: m = I + G/128, 6 WMMA squarings -> m^64, tr(m^128) via A.*A^T
  __syncthreads();
  for (int e = tid; e < 16384; e += 256) {
    int i = e >> 7, j = e & 127;
    float hard = (U[e] < gsoft[e]) ? 1.f : 0.f;
    M[e] = ((i == j) ? 1.f : 0.f) + hard * (1.f / 128.f);
  }
  __syncthreads();
  for (int s = 0; s < 6; ++s) square128_inplace(M);
  float tp = 0.f;
  for (int e = tid; e < 16384; e += 256)
    tp += M[e] * M[(e & 127) * 128 + (e >> 7)];
  float TR = block_reduce(tp, M);
  if (tid == 0) hk[k] = TR - 128.f;
}

// ---------------- Kernel 4: softmax weights + combined coeffs + sum(x^2) -----
__global__ void k_weights(const float* __restrict__ vals, const float* __restrict__ hk,
                          const float* __restrict__ xtx, float* __restrict__ wout,
                          float* __restrict__ cout, float* __restrict__ sumx2) {
  if (threadIdx.x == 0) {
    float mx = vals[0];
    for (int k = 1; k < KS; ++k) mx = fmaxf(mx, vals[k]);
    float tmp[KS]; float s = 0.f;
    for (int k = 0; k < KS; ++k) { tmp[k] = __expf(vals[k] - mx); s += tmp[k]; }
    float inv = 1.f / (s + 1e-30f);
    for (int k = 0; k < KS; ++k) {
      float wk = tmp[k] * inv;
      wout[k] = wk;
      cout[k] = wk - hk[k] * (1.f / 64.f);   // w_k - BETA*h_k/K
    }
    float sx = 0.f;
    for (int i = 0; i < 128; ++i) sx += xtx[i * 128 + i];
    *sumx2 = sx;
  }
}

// ---------------- Kernel 5: Dsum = sum_k c_k * (hard_k - g_soft) -------------
__global__ void k_dsum(const float* __restrict__ unif, const float* __restrict__ gsoft,
                       const float* __restrict__ c, float* __restrict__ dsum) {
  int e = blockIdx.x * 256 + threadIdx.x;
  float gs = gsoft[e];
  float s = 0.f;
  for (int k = 0; k < KS; ++k) {
    float hard = (unif[(size_t)k * 16384 + e] < gs) ? 1.f : 0.f;
    s += c[k] * (hard - gs);
  }
  dsum[e] = s;
}

// ---------------- Kernel 6: grad_z = -z + alpha*[Dsum@v ; Dsum^T@u] ----------
__global__ void k_gradz(const float* __restrict__ z, const float* __restrict__ dsum,
                        float* __restrict__ out) {
  int t = blockIdx.x * 256 + threadIdx.x;        // 4096 = 128*32
  int i = t >> 5, m = t & 31;
  float du = 0.f, dv = 0.f;
  for (int j = 0; j < 128; ++j) {
    du += dsum[i * 128 + j] * z[(j * KLV + m) * 2 + 1];   // diff @ v
    dv += dsum[j * 128 + i] * z[(j * KLV + m) * 2 + 0];   // diff^T @ u
  }
  int base = (i * KLV + m) * 2;
  out[base + 0] = -z[base + 0] + 0.1f * du;      // SIGMA_Z=1, ALPHA=0.1
  out[base + 1] = -z[base + 1] + 0.1f * dv;
}

// ---------------- Kernel 7: grad_theta = sum_k w_k grads_t[k] ----------------
__global__ void k_gradtheta(const float* __restrict__ grads, const float* __restrict__ wgt,
                            float* __restrict__ out) {
  int e = blockIdx.x * 256 + threadIdx.x;
  float s = 0.f;
  for (int k = 0; k < KS; ++k) s += wgt[k] * grads[(size_t)k * 16384 + e];
  out[8192 + e] = s;
}

// ---------------- Kernel 8: log_joint (soft graph) ---------------------------
__global__ void k_logjoint(const float* __restrict__ theta, const float* __restrict__ gsoft,
                           const float* __restrict__ xtx, const float* __restrict__ z,
                           const float* __restrict__ sumx2p, float* __restrict__ out) {
  __shared__ float M[16384];
  int tid = threadIdx.x, lane = tid & 31, w = tid >> 5;
  for (int e = tid; e < 16384; e += 256) M[e] = theta[e] * gsoft[e];   // Ws
  __syncthreads();
  float t1 = 0.f, t2 = 0.f, t3 = 0.f;
  int n0l = lane & 15, mb = (lane >> 4) ? 8 : 0;
  for (int nt = 0; nt < 8; ++nt) {
    v8f acc = v8f{0.f, 0.f, 0.f, 0.f, 0.f, 0.f, 0.f, 0.f};
    for (int k0 = 0; k0 < 128; k0 += 4) {
      v2f fa = loadA128(xtx, w, k0, lane);
      v2f fb = loadB128(M, nt, k0, lane);
      acc = wmma4(fa, fb, acc);
    }
#pragma unroll
    for (int r = 0; r < 8; ++r) {
      int e = (w * 16 + mb + r) * 128 + nt * 16 + n0l;
      float Ws = M[e];
      t1 += xtx[e] * Ws;
      t2 += Ws * acc[r];
      t3 += Ws * Ws;
    }
  }
  float zp = 0.f;
  for (int e = tid; e < 8192; e += 256) { float zz = z[e]; zp += zz * zz; }
  float T1 = block_reduce(t1, M);
  float T2 = block_reduce(t2, M);
  float T3 = block_reduce(t3, M);
  float TZ = block_reduce(zp, M);
  __syncthreads();
  for (int e = tid; e < 16384; e += 256) {
    int i = e >> 7, j = e & 127;
    M[e] = ((i == j) ? 1.f : 0.f) + gsoft[e] * (1.f / 128.f);
  }
  __syncthreads();
  for (int s = 0; s < 6; ++s) square128_inplace(M);
  float tp = 0.f;
  for (int e = tid; e < 16384; e += 256)
    tp += M[e] * M[(e & 127) * 128 + (e >> 7)];
  float TR = block_reduce(tp, M);
  if (tid == 0) {
    float sumx2 = *sumx2p;
    float ll  = (float)(NPTS * DD) * C_OBS - 0.5f * INV_S2 * (sumx2 - 2.f * T1 + T2);
    float lz  = (float)(DD * KLV * 2) * C_Z - 0.5f * TZ;
    float ltp = (float)(DD * DD) * C_TH - 0.5f * T3;
    float hs  = TR - 128.f;
    out[8192 + 16384] = ll + lz - hs + ltp;      // BETA=1
  }
}

extern "C" void kernel_launch(void* const* d_in, const int* in_sizes, int n_in,
                              void* d_out, int out_size, void* d_ws, size_t ws_size,
                              hipStream_t stream) {
  const float* z     = (const float*)d_in[0];   // (128,32,2)
  const float* theta = (const float*)d_in[1];   // (128,128)
  const float* x     = (const float*)d_in[2];   // (8192,128)
  const float* unif  = (const float*)d_in[3];   // (64,128,128)
  float* out = (float*)d_out;                   // 8192 + 16384 + 1 + 16384
  float* ws  = (float*)d_ws;

  float* gsoft = ws;                            // 16384
  float* xtx   = ws + 16384;                    // 16384
  float* grads = ws + 32768;                    // 64*16384
  float* vals  = grads + (size_t)KS * 16384;    // 64
  float* hk    = vals + 64;                     // 64
  float* wgt   = hk + 64;                       // 64
  float* cvec  = wgt + 64;                      // 64
  float* sumx2 = cvec + 64;                     // 1 (+pad)
  float* dsum  = sumx2 + 64;                    // 16384

  k_gsoft     <<<64, 256, 0, stream>>>(z, gsoft, out + 8192 + 16384 + 1);
  k_xtx       <<<64, 256, 0, stream>>>(x, xtx);
  k_perk      <<<64, 256, 0, stream>>>(theta, unif, gsoft, xtx, grads, vals, hk);
  k_weights   <<<1,  64, 0, stream>>>(vals, hk, xtx, wgt, cvec, sumx2);
  k_dsum      <<<64, 256, 0, stream>>>(unif, gsoft, cvec, dsum);
  k_gradz     <<<16, 256, 0, stream>>>(z, dsum, out);
  k_gradtheta <<<64, 256, 0, stream>>>(grads, wgt, out);
  k_logjoint  <<<1, 256, 0, stream>>>(theta, gsoft, xtx, z, sumx2, out);
}